// PaiNN_82695300317565
// MI455X (gfx1250) — compile-verified
//
#include <hip/hip_runtime.h>
#include <hip/hip_bf16.h>
#include <math.h>

// ---------------------------------------------------------------------------
// PaiNN forward for MI455X (gfx1250, wave32, WMMA)
// - phi = MLP(s_j) computed per-NODE (N=10k) instead of per-EDGE (~310k):
//   ~31x FLOP reduction; gathered in the edge kernel.
// - Dense GEMMs via V_WMMA_F32_16X16X4_F32 (full f32 accuracy; this workload
//   is scatter/latency bound, so low precision buys nothing).
// - Each wave drives TWO independent 16x16 accumulator chains sharing one
//   A fragment: 2x WMMA per LDS read, back-to-back independent WMMAs to
//   keep the XDL pipe busy (no D->A/B hazard between them).
// ---------------------------------------------------------------------------

typedef __attribute__((ext_vector_type(2))) float v2f;
typedef __attribute__((ext_vector_type(8))) float v8f;

#define F_DIM   128
#define F3      384
#define R_DIM   20
#define N_LAYERS 3
#define MAX_K   256
#define PI_OVER_CUT 0.6283185307179586f   /* pi / 5.0 */

__device__ __forceinline__ float silu_f(float x) {
    return x / (1.0f + expf(-x));
}

// ---------------------------------------------------------------------------
// 16-row-tile GEMM:  Out = act( A(16xK) @ W(KxN) + bias )
// Block = 128 threads = 4 wave32s; block covers 16 rows x 128 cols.
// Wave w owns column strips [w*16, w*16+16) and [64+w*16, 64+w*16+16)
// with two independent accumulators fed by one shared A fragment.
// A staged in LDS (pad +1 against bank conflicts); W streamed from L2.
// Strided A / strided Out so 'nfc,fg->ngc' einsums reuse this kernel.
// WMMA f32 16x16x4 layouts (ISA 7.12.2):
//   A: lane l, vgpr r -> A[l&15][k], k = 2*(l>>4)+r
//   B: lane l, vgpr r -> B[k][l&15], same k
//   C/D: vgpr p -> row p + 8*(l>>4), col l&15
// ---------------------------------------------------------------------------
__global__ __launch_bounds__(128)
void gemm16_wmma(const float* __restrict__ A, const float* __restrict__ W,
                 const float* __restrict__ bias, float* __restrict__ Out,
                 int K, int cols,
                 int lda, int kstride, int aoff,
                 int ldo, int ostride, int ooff,
                 int act)
{
    __shared__ float At[16 * (MAX_K + 1)];
    const int row0 = blockIdx.x << 4;
    const int ldk  = K + 1;

    // cooperative stage of the 16xK A tile
    for (int idx = threadIdx.x; idx < 16 * K; idx += blockDim.x) {
        int m = idx / K;
        int k = idx - m * K;
        At[m * ldk + k] = A[(size_t)(row0 + m) * lda + (size_t)k * kstride + aoff];
    }
    __syncthreads();

    const int lane    = threadIdx.x & 31;
    const int wave    = threadIdx.x >> 5;
    const int colbase = (blockIdx.y << 7) + (wave << 4);  // tile0; tile1 at +64
    const int mlo     = lane & 15;
    const int khi     = (lane >> 4) << 1;   // 0 or 2

    v8f acc0 = {0.f, 0.f, 0.f, 0.f, 0.f, 0.f, 0.f, 0.f};
    v8f acc1 = {0.f, 0.f, 0.f, 0.f, 0.f, 0.f, 0.f, 0.f};

    for (int kt = 0; kt < K; kt += 4) {
        const int k0 = kt + khi;
        v2f a, b0, b1;
        a.x = At[mlo * ldk + k0];
        a.y = At[mlo * ldk + k0 + 1];
        const float* wp = W + (size_t)k0 * cols + colbase + mlo;
        b0.x = wp[0];
        b0.y = wp[cols];
        b1.x = wp[64];
        b1.y = wp[cols + 64];
        acc0 = __builtin_amdgcn_wmma_f32_16x16x4_f32(
                   false, a, false, b0, (short)0, acc0, false, false);
        acc1 = __builtin_amdgcn_wmma_f32_16x16x4_f32(
                   false, a, false, b1, (short)0, acc1, false, false);
    }

    const int col0  = colbase + mlo;
    const int col1  = col0 + 64;
    const float bv0 = bias ? bias[col0] : 0.0f;
    const float bv1 = bias ? bias[col1] : 0.0f;
    const int rbase = row0 + ((lane >> 4) << 3);
#pragma unroll
    for (int p = 0; p < 8; ++p) {
        float x0 = acc0[p] + bv0;
        float x1 = acc1[p] + bv1;
        if (act) { x0 = silu_f(x0); x1 = silu_f(x1); }
        const size_t rb = (size_t)(rbase + p) * ldo;
        Out[rb + (size_t)col0 * ostride + ooff] = x0;
        Out[rb + (size_t)col1 * ostride + ooff] = x1;
    }
}

// ---------------------------------------------------------------------------
// s = emb[z], v = 0
// ---------------------------------------------------------------------------
__global__ void init_sv(const int* __restrict__ z, const float* __restrict__ emb,
                        float* __restrict__ s, float* __restrict__ v, int n_total)
{
    int idx = blockIdx.x * blockDim.x + threadIdx.x;
    if (idx >= n_total) return;
    int n = idx >> 7, f = idx & 127;
    s[idx] = emb[(size_t)z[n] * F_DIM + f];
    float* vp = v + (size_t)n * F3 + f * 3;
    vp[0] = 0.f; vp[1] = 0.f; vp[2] = 0.f;
}

// ---------------------------------------------------------------------------
// Per-edge geometry (computed once, reused for all layers)
// ---------------------------------------------------------------------------
__global__ void edge_geom(const int* __restrict__ ej, const int* __restrict__ ei,
                          int E, const float* __restrict__ pos,
                          float* __restrict__ rbfE, float* __restrict__ cutE,
                          float* __restrict__ rhatE)
{
    int e = blockIdx.x * blockDim.x + threadIdx.x;
    if (e >= E) return;
    int j = ej[e], i = ei[e];
    float rx = pos[i * 3 + 0] - pos[j * 3 + 0];
    float ry = pos[i * 3 + 1] - pos[j * 3 + 1];
    float rz = pos[i * 3 + 2] - pos[j * 3 + 2];
    float d   = sqrtf(rx * rx + ry * ry + rz * rz);
    float inv = 1.0f / fmaxf(d, 1e-12f);
    rhatE[(size_t)e * 3 + 0] = rx * inv;
    rhatE[(size_t)e * 3 + 1] = ry * inv;
    rhatE[(size_t)e * 3 + 2] = rz * inv;
    float dinv = (d == 0.0f) ? 1.0f : (1.0f / d);
#pragma unroll
    for (int r = 0; r < R_DIM; ++r)
        rbfE[(size_t)e * R_DIM + r] = sinf(d * (float)(r + 1) * PI_OVER_CUT) * dinv;
    cutE[e] = 0.5f * (cosf(d * PI_OVER_CUT) + 1.0f) * (d < 5.0f ? 1.0f : 0.0f);
}

// ---------------------------------------------------------------------------
// Edge message pass: one 128-thread block per edge, thread = feature f.
// Fuses Wt = (rbf @ W_rbf + b)*cut (K=20 FMAs in registers), gathers phi[j],
// v_prev[j]; scatters with float atomics into s[i], v[i].
// ---------------------------------------------------------------------------
__global__ __launch_bounds__(128)
void edge_message(const int* __restrict__ ej, const int* __restrict__ ei, int E,
                  const float* __restrict__ rbfE, const float* __restrict__ cutE,
                  const float* __restrict__ rhatE,
                  const float* __restrict__ phi,
                  const float* __restrict__ Wrbf, const float* __restrict__ brbf,
                  const float* __restrict__ vprev,
                  float* __restrict__ sacc, float* __restrict__ vacc)
{
    const int e = blockIdx.x;
    __shared__ float sh[24];   // [0..19]=rbf, [20..22]=rhat, [23]=cut
    if (threadIdx.x < R_DIM)
        sh[threadIdx.x] = rbfE[(size_t)e * R_DIM + threadIdx.x];
    else if (threadIdx.x < 23)
        sh[threadIdx.x] = rhatE[(size_t)e * 3 + (threadIdx.x - R_DIM)];
    else if (threadIdx.x == 23)
        sh[23] = cutE[e];
    __syncthreads();

    const int j = ej[e], i = ei[e];
    const int f = threadIdx.x;
    const float cut = sh[23];

    float wt0 = brbf[f], wt1 = brbf[F_DIM + f], wt2 = brbf[2 * F_DIM + f];
#pragma unroll
    for (int r = 0; r < R_DIM; ++r) {
        const float rb = sh[r];
        const float* wr = Wrbf + r * F3;
        wt0 += rb * wr[f];
        wt1 += rb * wr[F_DIM + f];
        wt2 += rb * wr[2 * F_DIM + f];
    }
    wt0 *= cut; wt1 *= cut; wt2 *= cut;

    const float* ph = phi + (size_t)j * F3;
    const float left  = ph[f]             * wt0;
    const float dsm   = ph[F_DIM + f]     * wt1;
    const float right = ph[2 * F_DIM + f] * wt2;

    atomicAdd(&sacc[(size_t)i * F_DIM + f], dsm);

    const float* vj = vprev + (size_t)j * F3 + f * 3;
    float*       vo = vacc  + (size_t)i * F3 + f * 3;
    atomicAdd(&vo[0], vj[0] * left + right * sh[20]);
    atomicAdd(&vo[1], vj[1] * left + right * sh[21]);
    atomicAdd(&vo[2], vj[2] * left + right * sh[22]);
}

// ---------------------------------------------------------------------------
// dot = sum_c Uv*Vv ; Vnorm = ||Vv|| ; cat = [s, Vnorm]
// ---------------------------------------------------------------------------
__global__ void dot_norm_cat(const float* __restrict__ Uv, const float* __restrict__ Vv,
                             const float* __restrict__ s,
                             float* __restrict__ dotb, float* __restrict__ cat,
                             int n_total)
{
    int idx = blockIdx.x * blockDim.x + threadIdx.x;
    if (idx >= n_total) return;
    int n = idx >> 7, g = idx & 127;
    const float* u = Uv + (size_t)n * F3 + g * 3;
    const float* w = Vv + (size_t)n * F3 + g * 3;
    dotb[idx] = u[0] * w[0] + u[1] * w[1] + u[2] * w[2];
    float nv  = sqrtf(w[0] * w[0] + w[1] * w[1] + w[2] * w[2]);
    cat[(size_t)n * 256 + g]         = s[idx];
    cat[(size_t)n * 256 + F_DIM + g] = nv;
}

// ---------------------------------------------------------------------------
// s += a_ss + a_sv*dot ; v += Uv * a_vv   (a = [a_vv | a_sv | a_ss])
// ---------------------------------------------------------------------------
__global__ void update_sv(const float* __restrict__ a, const float* __restrict__ dotb,
                          const float* __restrict__ Uv,
                          float* __restrict__ s, float* __restrict__ v, int n_total)
{
    int idx = blockIdx.x * blockDim.x + threadIdx.x;
    if (idx >= n_total) return;
    int n = idx >> 7, g = idx & 127;
    const float* an = a + (size_t)n * F3;
    const float avv = an[g], asv = an[F_DIM + g], ass = an[2 * F_DIM + g];
    s[idx] += ass + asv * dotb[idx];
    float*       vp = v  + (size_t)n * F3 + g * 3;
    const float* up = Uv + (size_t)n * F3 + g * 3;
    vp[0] += up[0] * avv;
    vp[1] += up[1] * avv;
    vp[2] += up[2] * avv;
}

// ---------------------------------------------------------------------------
extern "C" void kernel_launch(void* const* d_in, const int* in_sizes, int n_in,
                              void* d_out, int out_size, void* d_ws, size_t ws_size,
                              hipStream_t stream)
{
    const float* pos     = (const float*)d_in[0];
    const float* emb     = (const float*)d_in[1];
    const float* W_lin1  = (const float*)d_in[2];
    const float* b_lin1  = (const float*)d_in[3];
    const float* W_lin2m = (const float*)d_in[4];
    const float* b_lin2m = (const float*)d_in[5];
    const float* W_rbf   = (const float*)d_in[6];
    const float* b_rbf   = (const float*)d_in[7];
    const float* W_up    = (const float*)d_in[8];
    const float* b_up    = (const float*)d_in[9];
    const float* W_U     = (const float*)d_in[10];
    const float* W_V     = (const float*)d_in[11];
    const float* W_lin2u = (const float*)d_in[12];
    const float* b_lin2u = (const float*)d_in[13];
    const float* W_out   = (const float*)d_in[14];
    const float* b_out   = (const float*)d_in[15];
    const int*   z       = (const int*)d_in[16];
    const int*   eidx    = (const int*)d_in[17];

    const int Nn = in_sizes[16];          // 10000 (multiple of 16)
    const int E  = in_sizes[17] / 2;
    const int* ej = eidx;                 // edge_index[0] = source j
    const int* ei = eidx + E;             // edge_index[1] = target i

    // workspace carve-out
    float* ws = (float*)d_ws;
    size_t off = 0;
    auto carve = [&](size_t nelem) { float* p = ws + off; off += nelem; return p; };
    float* s     = carve((size_t)Nn * F_DIM);
    float* v     = carve((size_t)Nn * F3);
    float* vprev = carve((size_t)Nn * F3);
    float* mid   = carve((size_t)Nn * F_DIM);   // lin1 out / a1 / head tmp
    float* phi   = carve((size_t)Nn * F3);      // phi, reused as 'a'
    float* Uv    = carve((size_t)Nn * F3);
    float* Vv    = carve((size_t)Nn * F3);
    float* dotb  = carve((size_t)Nn * F_DIM);
    float* cat   = carve((size_t)Nn * 256);
    float* rbfE  = carve((size_t)E * R_DIM);
    float* cutE  = carve((size_t)E);
    float* rhatE = carve((size_t)E * 3);
    (void)ws_size; (void)n_in; (void)out_size;

    const int nt     = Nn * F_DIM;
    const int rtiles = Nn / 16;           // 625

    init_sv<<<(nt + 255) / 256, 256, 0, stream>>>(z, emb, s, v, nt);
    edge_geom<<<(E + 255) / 256, 256, 0, stream>>>(ej, ei, E, pos, rbfE, cutE, rhatE);

    for (int l = 0; l < N_LAYERS; ++l) {
        const float* wl1  = W_lin1  + (size_t)l * F_DIM * F_DIM;
        const float* bl1  = b_lin1  + (size_t)l * F_DIM;
        const float* wl2m = W_lin2m + (size_t)l * F_DIM * F3;
        const float* bl2m = b_lin2m + (size_t)l * F3;
        const float* wrb  = W_rbf   + (size_t)l * R_DIM * F3;
        const float* brb  = b_rbf   + (size_t)l * F3;
        const float* wup  = W_up    + (size_t)l * 256 * F_DIM;
        const float* bup  = b_up    + (size_t)l * F_DIM;
        const float* wU   = W_U     + (size_t)l * F_DIM * F_DIM;
        const float* wV   = W_V     + (size_t)l * F_DIM * F_DIM;
        const float* wl2u = W_lin2u + (size_t)l * F_DIM * F3;
        const float* bl2u = b_lin2u + (size_t)l * F3;

        // per-NODE message MLP: mid = silu(s @ W_lin1 + b); phi = mid @ W_lin2m + b
        gemm16_wmma<<<dim3(rtiles, 1), 128, 0, stream>>>(
            s, wl1, bl1, mid, F_DIM, F_DIM, F_DIM, 1, 0, F_DIM, 1, 0, 1);
        gemm16_wmma<<<dim3(rtiles, 3), 128, 0, stream>>>(
            mid, wl2m, bl2m, phi, F_DIM, F3, F_DIM, 1, 0, F3, 1, 0, 0);

        // snapshot v (edges must read pre-update v[j])
        hipMemcpyAsync(vprev, v, (size_t)Nn * F3 * sizeof(float),
                       hipMemcpyDeviceToDevice, stream);

        // edge scatter into s, v
        edge_message<<<E, 128, 0, stream>>>(ej, ei, E, rbfE, cutE, rhatE,
                                            phi, wrb, brb, vprev, s, v);

        // U_v / V_v : einsum 'nfc,fg->ngc' as 3 strided GEMMs each
        for (int c = 0; c < 3; ++c) {
            gemm16_wmma<<<dim3(rtiles, 1), 128, 0, stream>>>(
                v, wU, nullptr, Uv, F_DIM, F_DIM, F3, 3, c, F3, 3, c, 0);
            gemm16_wmma<<<dim3(rtiles, 1), 128, 0, stream>>>(
                v, wV, nullptr, Vv, F_DIM, F_DIM, F3, 3, c, F3, 3, c, 0);
        }

        dot_norm_cat<<<(nt + 255) / 256, 256, 0, stream>>>(Uv, Vv, s, dotb, cat, nt);

        // a = (silu(cat @ W_up + b_up)) @ W_lin2u + b_lin2u   (a reuses phi)
        gemm16_wmma<<<dim3(rtiles, 1), 128, 0, stream>>>(
            cat, wup, bup, mid, 256, F_DIM, 256, 1, 0, F_DIM, 1, 0, 1);
        gemm16_wmma<<<dim3(rtiles, 3), 128, 0, stream>>>(
            mid, wl2u, bl2u, phi, F_DIM, F3, F_DIM, 1, 0, F3, 1, 0, 0);

        update_sv<<<(nt + 255) / 256, 256, 0, stream>>>(phi, dotb, Uv, s, v, nt);
    }

    // output head: out = silu(s @ W_out + b_out) @ W_out + b_out
    gemm16_wmma<<<dim3(rtiles, 1), 128, 0, stream>>>(
        s, W_out, b_out, mid, F_DIM, F_DIM, F_DIM, 1, 0, F_DIM, 1, 0, 1);
    gemm16_wmma<<<dim3(rtiles, 1), 128, 0, stream>>>(
        mid, W_out, b_out, (float*)d_out, F_DIM, F_DIM, F_DIM, 1, 0, F_DIM, 1, 0, 0);
}